// Attention_62062277427791
// MI455X (gfx1250) — compile-verified
//
#include <hip/hip_runtime.h>
#include <cmath>

// ---------------- problem constants (from reference) ----------------
#define BATCH          64
#define NUM_HEADS      32
#define HEAD_DIM       128
#define NUM_KV         8
#define GROUP          4          // NUM_HEADS / NUM_KV
#define NUM_BLOCKS     4096
#define BLOCK_SZ       16
#define BLOCKS_PER_SEQ 64
#define SEQ_LEN        (BLOCKS_PER_SEQ * BLOCK_SZ)   // 1024
#define ATTN_SCALE     0.08838834764831845f
#define WAVES          4
#define TILES_PER_WAVE (BLOCKS_PER_SEQ / WAVES)      // 16
#define NSLOTS         (NUM_BLOCKS * BLOCK_SZ)       // 65536
#define VSTRIDE        132        // padded LDS row stride (floats) -> no bank conflicts

typedef __attribute__((ext_vector_type(2))) float v2f;
typedef __attribute__((ext_vector_type(8))) float v8f;

// ---------------- CDNA5 async VMEM->LDS helpers (inline asm path) ----------
// ISA: the immediate offset is added to both the LDS and the global address
__device__ __forceinline__ void async_load_b128_ofs(uint32_t lds_off,
                                                    const void* g, int ofs) {
  asm volatile("global_load_async_to_lds_b128 %0, %1, off offset:%2"
               :
               : "v"(lds_off), "v"(g), "i"(ofs)
               : "memory");
}
__device__ __forceinline__ void wait_asynccnt0() {
  asm volatile("s_wait_asynccnt 0x0" ::: "memory");
}
// Async loads complete in order: <=16 outstanding => previous tile's 16 landed.
__device__ __forceinline__ void wait_asynccnt_le16() {
  asm volatile("s_wait_asynccnt 0x10" ::: "memory");
}

// ---------------- prep: inverse slot map (avoids mutating k_cache/v_cache) --
__global__ void init_inv_kernel(int* __restrict__ inv) {
  int i = blockIdx.x * blockDim.x + threadIdx.x;
  if (i < NSLOTS) inv[i] = -1;
}

__global__ void fill_inv_kernel(const int* __restrict__ slot_mapping,
                                int* __restrict__ inv) {
  int i = threadIdx.x;
  if (i < BATCH) inv[slot_mapping[i]] = i;
}

// ---------------- paged-attention decode, fp32 WMMA 16x16x4 ----------------
// grid: (NUM_KV, BATCH), block: 128 threads = 4 wave32s.
// wave w handles tokens [w*256, (w+1)*256) = 16 page-tiles of 16 tokens.
// V tiles are double-buffered through LDS via async DMA.
__global__ __launch_bounds__(128) void paged_attn_kernel(
    const float* __restrict__ q,        // [B, H, D]
    const float* __restrict__ knew,     // [B, KV, D]
    const float* __restrict__ vnew,     // [B, KV, D]
    const float* __restrict__ k_cache,  // [NB, BS, KV, D]
    const float* __restrict__ v_cache,  // [NB, BS, KV, D]
    const int*   __restrict__ block_tables, // [B, BPS]
    const int*   __restrict__ inv,      // [NSLOTS]
    float*       __restrict__ out)      // [B, H*D]
{
  const int kv   = blockIdx.x;
  const int b    = blockIdx.y;
  const int tid  = threadIdx.x;
  const int wave = tid >> 5;
  const int lane = tid & 31;
  const bool lo  = (lane < 16);

  __shared__ float sm_m[WAVES][GROUP];
  __shared__ float sm_l[WAVES][GROUP];
  __shared__ float sm_o[WAVES][GROUP][HEAD_DIM];
  __shared__ float sm_v[WAVES][2][BLOCK_SZ * VSTRIDE];  // ping-pong V staging

  float* const vbufp[2] = {&sm_v[wave][0][0], &sm_v[wave][1][0]};
  const uint32_t vbuf_lds[2] = {(uint32_t)(uintptr_t)vbufp[0],
                                (uint32_t)(uintptr_t)vbufp[1]};

  const int tok = lane & 15;  // token within tile owned by this lane (A rows)

  // Issue 16 async B128 ops staging tile's 16x128 f32 V rows into LDS buf.
  // Lane l owns row l/2, half-row (l&1)*256B; IOFFSET walks both addresses.
  auto stage_v = [&](int tile, int buf) {
    const int pb2   = block_tables[b * BLOCKS_PER_SEQ + tile];
    const int slot2 = pb2 * BLOCK_SZ + tok;
    const int iv2   = inv[slot2];
    const float* vrow = (iv2 >= 0)
        ? (vnew    + ((size_t)iv2   * NUM_KV + kv) * HEAD_DIM)
        : (v_cache + ((size_t)slot2 * NUM_KV + kv) * HEAD_DIM);
    const uint32_t vlo = (uint32_t)(uintptr_t)vrow;
    const uint32_t vhi = (uint32_t)(((uintptr_t)vrow) >> 32);
    const uint32_t slo = __shfl(vlo, lane >> 1, 32);
    const uint32_t shi = __shfl(vhi, lane >> 1, 32);
    const char* grow = (const char*)(((uint64_t)shi << 32) | (uint64_t)slo)
                       + (lane & 1) * 256;
    const uint32_t lrow = vbuf_lds[buf]
                          + (uint32_t)(lane >> 1) * (VSTRIDE * 4)
                          + (uint32_t)(lane & 1) * 256;
#pragma unroll
    for (int i = 0; i < 16; ++i) {
      async_load_b128_ofs(lrow, grow, 16 * i);
    }
  };

  // ---- Q fragments (B matrix of QK^T): N column = lane%16, head replicated x4
  const int ghead = lane & 3;
  const float* qrow =
      q + ((size_t)b * NUM_HEADS + (size_t)kv * GROUP + ghead) * HEAD_DIM;
  const int hoff = (lane >> 4) << 1;  // +0 for lanes 0-15, +2 for lanes 16-31
  v2f qf[32];
#pragma unroll
  for (int c = 0; c < 32; ++c) {
    qf[c] = *(const v2f*)(qrow + 4 * c + hoff);
  }

  v8f oacc[8] = {};          // PV accumulators: rows = heads (replicated x4)
  float m_run = -INFINITY;   // per-head running max (head = lane%16 & 3)
  float l_run = 0.0f;        // per-head running sum

  const int tile0 = wave * TILES_PER_WAVE;
  stage_v(tile0, 0);         // prologue: DMA first V tile

  for (int t = 0; t < TILES_PER_WAVE; ++t) {
    const int tile = tile0 + t;
    const int pb   = block_tables[b * BLOCKS_PER_SEQ + tile];
    const int slot = pb * BLOCK_SZ + tok;
    const int iv   = inv[slot];
    const float* krow = (iv >= 0)
        ? (knew    + ((size_t)iv   * NUM_KV + kv) * HEAD_DIM)
        : (k_cache + ((size_t)slot * NUM_KV + kv) * HEAD_DIM);

    // ---- prefetch next tile's V via async DMA into the other buffer
    const bool have_next = (t + 1 < TILES_PER_WAVE);
    if (have_next) stage_v(tile + 1, (t + 1) & 1);

    // ---- scores: C[16 tok x 16 headcols] = K_tile x Q^T, 32 WMMAs over D.
    // Batch the 32 K-fragment loads so the scheduler can keep them in flight.
    v2f ka[32];
#pragma unroll
    for (int c = 0; c < 32; ++c) {
      ka[c] = *(const v2f*)(krow + 4 * c + hoff);
    }
    v8f sc = {};
#pragma unroll
    for (int c = 0; c < 32; ++c) {
      sc = __builtin_amdgcn_wmma_f32_16x16x4_f32(
          false, ka[c], false, qf[c], (short)0, sc, false, false);
    }

    // ---- online softmax (lane pair (l, l+16) holds all 16 tokens of a head)
    float s[8], p[8];
    float tmax = -INFINITY;
#pragma unroll
    for (int r = 0; r < 8; ++r) {
      s[r] = sc[r] * ATTN_SCALE;
      tmax = fmaxf(tmax, s[r]);
    }
    tmax = fmaxf(tmax, __shfl_xor(tmax, 16, 32));
    const float m_new = fmaxf(m_run, tmax);
    const float corr  = __expf(m_run - m_new);
    float psum = 0.0f;
#pragma unroll
    for (int r = 0; r < 8; ++r) {
      p[r] = __expf(s[r] - m_new);
      psum += p[r];
    }
    psum += __shfl_xor(psum, 16, 32);
    l_run = l_run * corr + psum;
    m_run = m_new;

    // rescale output accumulators: row v is head v&3; corr lives in lane (v&3)
    const float frow[4] = {__shfl(corr, 0, 32), __shfl(corr, 1, 32),
                           __shfl(corr, 2, 32), __shfl(corr, 3, 32)};
#pragma unroll
    for (int n = 0; n < 8; ++n) {
#pragma unroll
      for (int v = 0; v < 8; ++v) oacc[n][v] *= frow[v & 3];
    }

    // ---- this tile's V DMA has landed once <=16 (next tile's) remain
    if (have_next) wait_asynccnt_le16();
    else           wait_asynccnt0();

    // ---- PV: O[16 heads x 128 dims] += P x V_tile (8 dim-tiles x 4 chunks)
    const float* vbuf = vbufp[t & 1];
    const int dcol = lane & 15;
#pragma unroll
    for (int cc = 0; cc < 4; ++cc) {
      // re-fragment P from C layout into A layout via lane-pair swaps
      v2f pa;
      if (cc < 2) {  // tokens 0..7 region live in low lanes
        const float rx = __shfl_xor(p[4 * cc + 2], 16, 32);
        const float ry = __shfl_xor(p[4 * cc + 3], 16, 32);
        pa.x = lo ? p[4 * cc]     : rx;
        pa.y = lo ? p[4 * cc + 1] : ry;
      } else {       // tokens 8..15 region live in high lanes
        const float rx = __shfl_xor(p[4 * cc - 8], 16, 32);
        const float ry = __shfl_xor(p[4 * cc - 7], 16, 32);
        pa.x = lo ? rx : p[4 * cc - 6];
        pa.y = lo ? ry : p[4 * cc - 5];
      }
      const int tok0 = 4 * cc + (lo ? 0 : 2);  // B-frag rows for this lane half
#pragma unroll
      for (int n = 0; n < 8; ++n) {
        v2f vb;
        vb.x = vbuf[tok0 * VSTRIDE + 16 * n + dcol];
        vb.y = vbuf[(tok0 + 1) * VSTRIDE + 16 * n + dcol];
        oacc[n] = __builtin_amdgcn_wmma_f32_16x16x4_f32(
            false, pa, false, vb, (short)0, oacc[n], false, false);
      }
    }
  }

  // ---- per-wave partials to LDS
  if (lane < GROUP) {
    sm_m[wave][lane] = m_run;
    sm_l[wave][lane] = l_run;
  }
  if (lo) {
#pragma unroll
    for (int n = 0; n < 8; ++n) {
#pragma unroll
      for (int v = 0; v < GROUP; ++v) {
        sm_o[wave][v][16 * n + lane] = oacc[n][v];
      }
    }
  }
  __syncthreads();

  // ---- split-S flash combine across the 4 waves
  float* outb = out + ((size_t)b * NUM_HEADS + (size_t)kv * GROUP) * HEAD_DIM;
  for (int e = tid; e < GROUP * HEAD_DIM; e += 128) {
    const int g = e >> 7;
    const int d = e & (HEAD_DIM - 1);
    float M = sm_m[0][g];
#pragma unroll
    for (int w = 1; w < WAVES; ++w) M = fmaxf(M, sm_m[w][g]);
    float L = 0.0f, O = 0.0f;
#pragma unroll
    for (int w = 0; w < WAVES; ++w) {
      const float f = __expf(sm_m[w][g] - M);
      L += sm_l[w][g] * f;
      O += sm_o[w][g][d] * f;
    }
    outb[(size_t)g * HEAD_DIM + d] = O / L;
  }
}

// ---------------- launch ----------------
extern "C" void kernel_launch(void* const* d_in, const int* in_sizes, int n_in,
                              void* d_out, int out_size, void* d_ws, size_t ws_size,
                              hipStream_t stream) {
  const float* q    = (const float*)d_in[0];
  const float* knew = (const float*)d_in[1];
  const float* vnew = (const float*)d_in[2];
  const float* kc   = (const float*)d_in[3];
  const float* vc   = (const float*)d_in[4];
  const int*   slot = (const int*)d_in[5];
  const int*   bt   = (const int*)d_in[6];
  float*       out  = (float*)d_out;
  int*         inv  = (int*)d_ws;   // NSLOTS ints = 256 KB of workspace

  init_inv_kernel<<<(NSLOTS + 255) / 256, 256, 0, stream>>>(inv);
  fill_inv_kernel<<<1, BATCH, 0, stream>>>(slot, inv);
  paged_attn_kernel<<<dim3(NUM_KV, BATCH), 128, 0, stream>>>(
      q, knew, vnew, kc, vc, bt, inv, out);
}